// SoftSODA_70506183131678
// MI455X (gfx1250) — compile-verified
//
#include <hip/hip_runtime.h>

// Soft-SODA DP:  R[i,j] = softmin_1( R[i-1,j-1]+D[i-1,j-1], R[i-1,j], R[i,j-1] )
// Anti-diagonal wavefront; thread t owns row i=t+1.  Neighbor values travel in
// registers (shfl_up inside a wave); only wave boundaries go through LDS
// (parity double-buffered b64 slots, one split-barrier per step).  Slot 0 is a
// permanent zero so wave 0 has no boundary special case.  The step body is
// manually expanded exactly twice (parities as literals) and the outer loop is
// pinned with unroll(1) so the backend cannot explode the code size.

#define SODA_N 1024
#define SODA_M 1024
#define NWAVES 32

// One DP step.  RP = parity of the slot row to read ((d-1)&1),
//               WP = parity of the slot row to write (d&1).
#define SODA_STEP(RP, WP)                                                     \
    do {                                                                      \
        const bool valid = ((unsigned)jj) < (unsigned)SODA_M;                 \
        float cost = 0.0f;                                                    \
        const int ph = jj & 3;                                                \
        if (valid && ph == 0) {                                               \
            cur = nxt;                              /* consume staged chunk */\
            if (jj + 4 < SODA_M) { nxt = *nxtp; ++nxtp; }                     \
        }                                                                     \
        if (valid)                                                            \
            cost = (ph == 0) ? cur.x : (ph == 1) ? cur.y                      \
                 : (ph == 2) ? cur.z : cur.w;                                 \
        float nb1 = __shfl_up(v1, 1, 32);           /* R[i-1, j]   */         \
        float nb2 = __shfl_up(v2, 1, 32);           /* R[i-1, j-1] */         \
        const float2 pz = slots[RP][wv];            /* uniform broadcast */   \
        nb1 = isl0 ? pz.x : nb1;                                              \
        nb2 = isl0 ? pz.y : nb2;                                              \
        const float a  = nb2 + cost;                                          \
        const float bu = nb1;                                                 \
        const float cl = v1;                                                  \
        const float p  = fminf(a, bu);                                        \
        const float q  = fmaxf(a, bu);                                        \
        const float m  = fminf(p, cl);              /* min3    */             \
        const float mx = fmaxf(q, cl);              /* max3    */             \
        const float md = fmaxf(p, fminf(q, cl));    /* median3 */             \
        const float s  = 1.0f + __expf(m - md) + __expf(m - mx);              \
        const float val = m - __logf(s);                                      \
        v2 = v1;                                                              \
        v1 = valid ? val : 0.0f;                                              \
        if (isl31) slots[WP][wv + 1] = make_float2(v1, v2);                   \
        __syncthreads();                                                      \
        ++jj;                                                                 \
    } while (0)

__global__ __launch_bounds__(1024, 1)
void soda_dp_kernel(const float* __restrict__ D, float* __restrict__ out) {
    const int b    = blockIdx.x;
    const int tid  = threadIdx.x;         // 0..1023
    const int i    = tid + 1;             // DP row 1..N (fixed per thread)
    const int lane = tid & 31;
    const int wv   = tid >> 5;
    const bool isl0  = (lane == 0);
    const bool isl31 = (lane == 31);

    const float* __restrict__ row =
        D + (size_t)b * SODA_N * SODA_M + (size_t)(i - 1) * SODA_M;

    // Warm L2 with this thread's whole cost row up front (global_prefetch_b8,
    // 128B cacheline stride).  All of D (64MB) fits in the 192MB L2.
    #pragma unroll
    for (int k = 0; k < SODA_M; k += 32)
        __builtin_prefetch(row + k, 0, 0);

    // slots[parity][w] = (v1,v2) of wave w-1's lane 31 after the step of that
    // parity; index 0 is never written -> permanent zero boundary for wave 0.
    __shared__ float2 slots[2][NWAVES + 1];
    if (tid < 2 * (NWAVES + 1)) {
        const int p = tid / (NWAVES + 1);
        const int s = tid - p * (NWAVES + 1);
        slots[p][s] = make_float2(0.0f, 0.0f);
    }
    __syncthreads();

    // Register state: v1 = my (masked) value on diagonal d-1,
    //                 v2 = my (masked) value on diagonal d-2.
    float v1 = 0.0f, v2 = 0.0f;

    // Register double-buffered row streaming: one b128 load per 4 DP steps.
    float4 cur = make_float4(0.f, 0.f, 0.f, 0.f);
    float4 nxt = *reinterpret_cast<const float4*>(row);      // columns 0..3
    const float4* nxtp = reinterpret_cast<const float4*>(row) + 1;

    int jj = 1 - i;                        // cost column index at d = 2

    // d = 2..2048 : 1023 x (even step, odd step) + one final even step.
    #pragma unroll 1
    for (int it = 0; it < (SODA_N + SODA_M - 1) / 2; ++it) {
        SODA_STEP(1, 0);                   // even d: read slots[1], write [0]
        SODA_STEP(0, 1);                   // odd  d: read slots[0], write [1]
    }
    SODA_STEP(1, 0);                       // d = N+M (even)

    // Thread owning row N: its final (valid) step was d = N+M -> v1 = R[N,M].
    if (tid == SODA_N - 1) out[b] = -v1;
}

extern "C" void kernel_launch(void* const* d_in, const int* in_sizes, int n_in,
                              void* d_out, int out_size, void* d_ws, size_t ws_size,
                              hipStream_t stream) {
    (void)n_in; (void)d_ws; (void)ws_size; (void)out_size;
    const float* D   = (const float*)d_in[0];
    float*       out = (float*)d_out;

    const int B = in_sizes[0] / (SODA_N * SODA_M);           // 16 here
    soda_dp_kernel<<<dim3(B), dim3(1024), 0, stream>>>(D, out);
}